// HierarchicalOutputLayer_44229573214687
// MI455X (gfx1250) — compile-verified
//
#include <hip/hip_runtime.h>
#include <hip/hip_bf16.h>

typedef __attribute__((ext_vector_type(2))) float v2f;
typedef __attribute__((ext_vector_type(8))) float v8f;

#define D_DIM   512
#define T_LEN   4096
#define NOTE    8192
#define KPAIRS  256            // D_DIM / 2
#define TILE_F2 (KPAIRS * 16)  // float2 elements per packed N-tile (== 4096)

// ---------------------------------------------------------------------------
// Weight packing: Wcat[512][48] (cols 0-4 Wg, 5-14 Wt, 15 zero, 16-40 Wf,
// 41-47 zero) stored as [tile][k_pair][n] float2 so a B-fragment lane load
// is a single contiguous 8-byte load.
// ---------------------------------------------------------------------------
__device__ __forceinline__ float wcat_elem(const float* Wg, const float* Wt,
                                           const float* Wf, int d, int c) {
  if (c < 5)  return Wg[d * 5 + c];
  if (c < 15) return Wt[d * 10 + (c - 5)];
  if (c < 16) return 0.0f;
  if (c < 41) return Wf[d * 25 + (c - 16)];
  return 0.0f;
}

__global__ void pack_weights_kernel(const float* __restrict__ Wg,
                                    const float* __restrict__ Wt,
                                    const float* __restrict__ Wf,
                                    float* __restrict__ Wpack) {
  int idx = blockIdx.x * blockDim.x + threadIdx.x;  // one float2 slot each
  if (idx >= 3 * TILE_F2) return;
  int n    = idx & 15;
  int kk   = (idx >> 4) & (KPAIRS - 1);
  int tile = idx >> 12;                 // TILE_F2 == 2^12
  int col  = tile * 16 + n;
  Wpack[2 * idx]     = wcat_elem(Wg, Wt, Wf, 2 * kk,     col);
  Wpack[2 * idx + 1] = wcat_elem(Wg, Wt, Wf, 2 * kk + 1, col);
}

// ---------------------------------------------------------------------------
// Main kernel: each wave computes a 16-row x 48-col tile of x @ Wcat with
// V_WMMA_F32_16X16X4_F32, then one lane per row does the hierarchical
// argmax/mask cascade and writes frame logits + 2x-upsampled outputs.
// ---------------------------------------------------------------------------
__global__ __launch_bounds__(128) void hier_wmma_kernel(
    const float* __restrict__ x, const float* __restrict__ Wpack,
    const float* __restrict__ bg, const float* __restrict__ bt,
    const float* __restrict__ bf,
    float* __restrict__ group_up, float* __restrict__ tech_up,
    float* __restrict__ final_up, float* __restrict__ frame_out) {
  __shared__ float sm[4][16][52];

  const int wave = threadIdx.x >> 5;      // wave32
  const int lane = threadIdx.x & 31;
  const int tile = blockIdx.x * 4 + wave; // 4096 row-tiles total
  const int row0 = tile << 4;
  const int m    = lane & 15;
  const int half = lane >> 4;             // K half-wave split (A & B frags)
  const int n    = lane & 15;

  // A fragment: lane holds x[row0+m, k0 + 2*half .. +1]  (8B contiguous)
  const float* xrow = x + (size_t)(row0 + m) * D_DIM + 2 * half;
  // B fragment: lane holds Wcat[k0 + 2*half .. +1][tile*16 + n] as one float2
  const v2f* Wp  = (const v2f*)Wpack;
  const v2f* wp0 = Wp + 0 * TILE_F2 + half * 16 + n;
  const v2f* wp1 = Wp + 1 * TILE_F2 + half * 16 + n;
  const v2f* wp2 = Wp + 2 * TILE_F2 + half * 16 + n;

  v8f acc0 = {}, acc1 = {}, acc2 = {};

#pragma unroll 4
  for (int k0 = 0; k0 < D_DIM; k0 += 4) {
    v2f a  = *(const v2f*)(xrow + k0);
    int po = (k0 >> 1) << 4;  // (k0/2) pairs * 16 cols
    v2f b0 = wp0[po];
    v2f b1 = wp1[po];
    v2f b2 = wp2[po];
    acc0 = __builtin_amdgcn_wmma_f32_16x16x4_f32(false, a, false, b0,
                                                 (short)0, acc0, false, false);
    acc1 = __builtin_amdgcn_wmma_f32_16x16x4_f32(false, a, false, b1,
                                                 (short)0, acc1, false, false);
    acc2 = __builtin_amdgcn_wmma_f32_16x16x4_f32(false, a, false, b2,
                                                 (short)0, acc2, false, false);
  }

  // C/D layout: VGPR r, lanes 0-15 -> (M=r, N=lane), lanes 16-31 -> (M=r+8).
  {
    const int rbase = half << 3;
#pragma unroll
    for (int r = 0; r < 8; ++r) {
      sm[wave][rbase + r][n]      = acc0[r];
      sm[wave][rbase + r][16 + n] = acc1[r];
      sm[wave][rbase + r][32 + n] = acc2[r];
    }
  }
  __syncthreads();

  if (lane < 16) {
    const float* rowv = sm[wave][lane];

    // --- group level (5) ---
    float g[5];
#pragma unroll
    for (int i = 0; i < 5; ++i) g[i] = rowv[i] + bg[i];
    float gbest = g[0]; int gi = 0;
#pragma unroll
    for (int i = 1; i < 5; ++i) if (g[i] > gbest) { gbest = g[i]; gi = i; }

    // --- tech level (10), masked by group ownership ---
    const int TOWN[10] = {0, 1, 2, 2, 2, 3, 3, 3, 4, 4};
    float tl[10];
#pragma unroll
    for (int i = 0; i < 10; ++i) {
      float msk = (gi == TOWN[i]) ? 1.0f : 0.0f;
      tl[i] = (rowv[5 + i] + bt[i]) * msk;
    }
    float tbest = tl[0]; int ti = 0;
#pragma unroll
    for (int i = 1; i < 10; ++i) if (tl[i] > tbest) { tbest = tl[i]; ti = i; }

    // --- final level (25), masked by tech ownership ---
    const int FOWN[25] = {0, 1, 2, 2, 2, 3, 3, 3, 3, 3, 3, 3, 3,
                          4, 9, 6, 8, 5, 7, 1, 3, 3, 3, 9, 3};
    float fl[25];
#pragma unroll
    for (int i = 0; i < 25; ++i) {
      float msk = (ti == FOWN[i]) ? 1.0f : 0.0f;
      fl[i] = (rowv[16 + i] + bf[i]) * msk;
    }

    // --- writes: frame logits + nearest-upsample (each frame -> 2 rows) ---
    const int R = row0 + lane;          // 0..65535
    const int b = R >> 12;              // / 4096
    const int t = R & (T_LEN - 1);
    const size_t u = (size_t)b * NOTE + 2 * (size_t)t;

    float* gp = group_up + u * 5;
#pragma unroll
    for (int i = 0; i < 5; ++i)  { gp[i] = g[i];  gp[5 + i]  = g[i];  }
    float* tp = tech_up + u * 10;
#pragma unroll
    for (int i = 0; i < 10; ++i) { tp[i] = tl[i]; tp[10 + i] = tl[i]; }
    float* fp = final_up + u * 25;
#pragma unroll
    for (int i = 0; i < 25; ++i) { fp[i] = fl[i]; fp[25 + i] = fl[i]; }
    float* fr = frame_out + (size_t)R * 25;
#pragma unroll
    for (int i = 0; i < 25; ++i) fr[i] = fl[i];
  }
}

// ---------------------------------------------------------------------------
extern "C" void kernel_launch(void* const* d_in, const int* in_sizes, int n_in,
                              void* d_out, int out_size, void* d_ws,
                              size_t ws_size, hipStream_t stream) {
  const float* x  = (const float*)d_in[0];
  const float* Wg = (const float*)d_in[1];
  const float* bg = (const float*)d_in[2];
  const float* Wt = (const float*)d_in[3];
  const float* bt = (const float*)d_in[4];
  const float* Wf = (const float*)d_in[5];
  const float* bf = (const float*)d_in[6];
  // d_in[7] = note_size (fixed 8192 for these shapes)

  float* out      = (float*)d_out;
  float* group_up = out;                                   // [16,8192,5]
  float* tech_up  = group_up + (size_t)16 * NOTE * 5;      // [16,8192,10]
  float* final_up = tech_up  + (size_t)16 * NOTE * 10;     // [16,8192,25]
  float* frame    = final_up + (size_t)16 * NOTE * 25;     // [16,4096,25]

  float* Wpack = (float*)d_ws;  // 3 * 512 * 16 floats = 96 KB

  pack_weights_kernel<<<48, 256, 0, stream>>>(Wg, Wt, Wf, Wpack);
  hier_wmma_kernel<<<1024, 128, 0, stream>>>(x, Wpack, bg, bt, bf,
                                             group_up, tech_up, final_up,
                                             frame);
}